// SinkhornSolver_1864015807159
// MI455X (gfx1250) — compile-verified
//
#include <hip/hip_runtime.h>
#include <math.h>

// ---------------------------------------------------------------------------
// Sinkhorn solver for MI455X (gfx1250).
//  * C (64MB) is L2-resident (192MB L2) -> passes are L2-BW / v_exp bound.
//  * u_new_i = eps*(log_a - LSE_j((v_j - C_ij)/eps))   (the +u term cancels)
//  * v_new_j = eps*(log_b - LSE_i((u_new_i - C_ij)/eps))
//  * Online max-stabilized LSE in base-2 (native v_exp_f32 / v_log_f32).
//  * Row pass stages each row through LDS with the CDNA5 async copy engine
//    (global_load_async_to_lds_b128, s_wait_asynccnt), double buffered.
//  * 3 kernels per iteration (launch overhead matters: each pass is only a
//    few microseconds of L2 traffic + v_exp work).
//  * Deterministic: no atomics; diff reduced with a fixed-order tree.
// ---------------------------------------------------------------------------

#define NDIM      4096
#define EPS       0.01f
#define KSCALE    144.2695040888963f    /* (1/eps) * log2(e) */
#define LN2F      0.6931471805599453f
#define LOG_AB    -8.317725207562705f   /* log(1/4096 + 1e-8), a==b */
#define THRESH    0.001f
#define ITERS     100
#define NSLICE    16                    /* row slices for the column pass */

// ------------------------- init: u = v = 0, done = 0 -----------------------
__global__ __launch_bounds__(256) void sk_init(float* __restrict__ u,
                                               float* __restrict__ v,
                                               int* __restrict__ done) {
  int i = blockIdx.x * 256 + threadIdx.x;
  u[i] = 0.0f;
  v[i] = 0.0f;
  if (i == 0) *done = 0;
}

// ---------------- row pass: one wave32 per row, async-LDS staged -----------
__global__ __launch_bounds__(256) void sk_row(const float* __restrict__ C,
                                              const float* __restrict__ u,
                                              const float* __restrict__ v,
                                              float* __restrict__ u_new,
                                              float* __restrict__ du) {
  __shared__ float4 stage[8][2][32];            // 8 waves x double buffer x 512B
  const int tid  = threadIdx.x;
  const int wave = tid >> 5;
  const int lane = tid & 31;
  const int row  = blockIdx.x * 8 + wave;

  const float4* gRow = (const float4*)(C + (size_t)row * NDIM); // 1024 quads
  const float4* v4   = (const float4*)v;

  // generic shared pointer low 32 bits == LDS byte offset (ISA 10.2)
  unsigned ldsA = (unsigned)(size_t)&stage[wave][0][lane];
  unsigned ldsB = (unsigned)(size_t)&stage[wave][1][lane];

  // prime buffer 0 with chunk 0 (each lane copies 16B; wave moves 512B)
  asm volatile("global_load_async_to_lds_b128 %0, %1, off"
               :: "v"(ldsA), "v"(gRow + lane) : "memory");

  float m = -3.0e38f, s = 0.0f;
  for (int c = 0; c < 32; ++c) {
    if (c + 1 < 32) {
      unsigned dst = ((c + 1) & 1) ? ldsB : ldsA;
      // make sure our DS reads of the buffer being overwritten retired
      asm volatile("s_wait_dscnt 0" ::: "memory");
      asm volatile("global_load_async_to_lds_b128 %0, %1, off"
                   :: "v"(dst), "v"(gRow + (c + 1) * 32 + lane) : "memory");
      asm volatile("s_wait_asynccnt 1" ::: "memory");   // chunk c landed
    } else {
      asm volatile("s_wait_asynccnt 0" ::: "memory");
    }
    float4 cv = stage[wave][c & 1][lane];
    float4 vv = v4[c * 32 + lane];
    float t0 = (vv.x - cv.x) * KSCALE;
    float t1 = (vv.y - cv.y) * KSCALE;
    float t2 = (vv.z - cv.z) * KSCALE;
    float t3 = (vv.w - cv.w) * KSCALE;
    float tm = fmaxf(fmaxf(t0, t1), fmaxf(t2, t3));
    float mn = fmaxf(m, tm);
    s = s * exp2f(m - mn)
        + exp2f(t0 - mn) + exp2f(t1 - mn)
        + exp2f(t2 - mn) + exp2f(t3 - mn);
    m = mn;
  }
  // wave32 (m,s) merge
  for (int off = 16; off >= 1; off >>= 1) {
    float mo = __shfl_xor(m, off, 32);
    float so = __shfl_xor(s, off, 32);
    float mn = fmaxf(m, mo);
    s = s * exp2f(m - mn) + so * exp2f(mo - mn);
    m = mn;
  }
  if (lane == 0) {
    float L  = (m + log2f(s)) * LN2F;           // natural-log LSE of the row
    float un = EPS * (LOG_AB - L);
    u_new[row] = un;
    du[row]    = fabsf(un - u[row]);            // vs pre-commit u (ref order)
  }
}

// ------------- column pass, stage 1: per-slice partial (m, s) --------------
__global__ __launch_bounds__(256) void sk_col_partial(const float* __restrict__ C,
                                                      const float* __restrict__ un,
                                                      float* __restrict__ pm,
                                                      float* __restrict__ ps) {
  const int j     = (blockIdx.x & 15) * 256 + threadIdx.x;  // column
  const int slice = blockIdx.x >> 4;                        // 0..NSLICE-1
  const int i0    = slice * (NDIM / NSLICE);
  const float* Cp = C + j;
  float m = -3.0e38f, s = 0.0f;
#pragma unroll 2
  for (int i = i0; i < i0 + (NDIM / NSLICE); i += 4) {
    float c0 = Cp[(size_t)(i + 0) * NDIM];      // coalesced across threads
    float c1 = Cp[(size_t)(i + 1) * NDIM];
    float c2 = Cp[(size_t)(i + 2) * NDIM];
    float c3 = Cp[(size_t)(i + 3) * NDIM];
    float t0 = (un[i + 0] - c0) * KSCALE;       // un[i] uniform -> s_load
    float t1 = (un[i + 1] - c1) * KSCALE;
    float t2 = (un[i + 2] - c2) * KSCALE;
    float t3 = (un[i + 3] - c3) * KSCALE;
    float tm = fmaxf(fmaxf(t0, t1), fmaxf(t2, t3));
    float mn = fmaxf(m, tm);
    s = s * exp2f(m - mn)
        + exp2f(t0 - mn) + exp2f(t1 - mn)
        + exp2f(t2 - mn) + exp2f(t3 - mn);
    m = mn;
  }
  pm[slice * NDIM + j] = m;
  ps[slice * NDIM + j] = s;
}

// -- column finish: combine slices -> v_new (regs), commit u/v, diff, done --
__global__ __launch_bounds__(1024) void sk_col_finish(const float* __restrict__ pm,
                                                      const float* __restrict__ ps,
                                                      const float* __restrict__ un,
                                                      const float* __restrict__ du,
                                                      float* __restrict__ u,
                                                      float* __restrict__ v,
                                                      int* __restrict__ done) {
  __shared__ float red[1024];
  const int t  = threadIdx.x;
  const int dn = *done;                 // pre-update flag (reference ordering)
  const int j0 = t * 4;                 // 4 columns per thread

  float m0 = -3.0e38f, m1 = -3.0e38f, m2 = -3.0e38f, m3 = -3.0e38f;
  float s0 = 0.f, s1 = 0.f, s2 = 0.f, s3 = 0.f;
#pragma unroll
  for (int k = 0; k < NSLICE; ++k) {
    float4 mo = *(const float4*)(pm + k * NDIM + j0);
    float4 so = *(const float4*)(ps + k * NDIM + j0);
    float n;
    n = fmaxf(m0, mo.x); s0 = s0 * exp2f(m0 - n) + so.x * exp2f(mo.x - n); m0 = n;
    n = fmaxf(m1, mo.y); s1 = s1 * exp2f(m1 - n) + so.y * exp2f(mo.y - n); m1 = n;
    n = fmaxf(m2, mo.z); s2 = s2 * exp2f(m2 - n) + so.z * exp2f(mo.z - n); m2 = n;
    n = fmaxf(m3, mo.w); s3 = s3 * exp2f(m3 - n) + so.w * exp2f(mo.w - n); m3 = n;
  }
  float4 vn;
  vn.x = EPS * (LOG_AB - (m0 + log2f(s0)) * LN2F);
  vn.y = EPS * (LOG_AB - (m1 + log2f(s1)) * LN2F);
  vn.z = EPS * (LOG_AB - (m2 + log2f(s2)) * LN2F);
  vn.w = EPS * (LOG_AB - (m3 + log2f(s3)) * LN2F);

  float4 vold = *(const float4*)(v + j0);       // v entering this iteration
  float4 duq  = *(const float4*)(du + j0);      // |u_new - u_old| from sk_row
  float acc = fabsf(vn.x - vold.x) + fabsf(vn.y - vold.y)
            + fabsf(vn.z - vold.z) + fabsf(vn.w - vold.w)
            + duq.x + duq.y + duq.z + duq.w;

  if (!dn) {                                    // freeze: u,v = where(done,...)
    float4 unq = *(const float4*)(un + j0);
    *(float4*)(v + j0) = vn;
    *(float4*)(u + j0) = unq;
  }

  red[t] = acc;                                 // deterministic tree reduction
  __syncthreads();
  for (int off = 512; off >= 1; off >>= 1) {
    if (t < off) red[t] += red[t + off];
    __syncthreads();
  }
  if (t == 0 && !dn && red[0] < THRESH) *done = 1;   // done |= diff < thr
}

// --------------------- final transport plan: pi = exp(K) -------------------
__global__ __launch_bounds__(256) void sk_pi(const float* __restrict__ C,
                                             const float* __restrict__ u,
                                             const float* __restrict__ v,
                                             float* __restrict__ out) {
  const size_t q = (size_t)blockIdx.x * 256 + threadIdx.x;  // quad index
  const size_t e = q * 4;
  const int i = (int)(e >> 12);
  const int j = (int)(e & (NDIM - 1));
  const float ui = u[i];
  float4 cv = *(const float4*)(C + e);
  float4 vv = *(const float4*)(v + j);
  float4 o;
  o.x = exp2f((ui + vv.x - cv.x) * KSCALE);
  o.y = exp2f((ui + vv.y - cv.y) * KSCALE);
  o.z = exp2f((ui + vv.z - cv.z) * KSCALE);
  o.w = exp2f((ui + vv.w - cv.w) * KSCALE);
  *(float4*)(out + e) = o;
}

// ---------------------------------------------------------------------------
extern "C" void kernel_launch(void* const* d_in, const int* in_sizes, int n_in,
                              void* d_out, int out_size, void* d_ws, size_t ws_size,
                              hipStream_t stream) {
  (void)in_sizes; (void)n_in; (void)out_size; (void)ws_size;
  const float* C = (const float*)d_in[2];   // inputs: x, y, C (only C is used)

  float* ws = (float*)d_ws;                 // workspace layout (floats):
  float* u  = ws + 0;                       // 4096
  float* v  = ws + 4096;                    // 4096
  float* un = ws + 8192;                    // 4096
  float* du = ws + 12288;                   // 4096
  float* pm = ws + 16384;                   // NSLICE*4096
  float* ps = ws + 16384 + NSLICE * NDIM;   // NSLICE*4096
  int* done = (int*)(ws + 16384 + 2 * NSLICE * NDIM);  // ~590KB total

  sk_init<<<16, 256, 0, stream>>>(u, v, done);
  for (int it = 0; it < ITERS; ++it) {
    sk_row        <<<512,         256, 0, stream>>>(C, u, v, un, du);
    sk_col_partial<<<16 * NSLICE, 256, 0, stream>>>(C, un, pm, ps);
    sk_col_finish <<<1,          1024, 0, stream>>>(pm, ps, un, du, u, v, done);
  }
  sk_pi<<<(NDIM / 4) * (NDIM / 256), 256, 0, stream>>>(C, u, v, (float*)d_out);
}